// EdgeAwareLoss_10479720202688
// MI455X (gfx1250) — compile-verified
//
#include <hip/hip_runtime.h>
#include <hip/hip_bf16.h>
#include <stdint.h>
#include <limits.h>

// Problem constants from the reference
#define B_  8
#define C_  2
#define H_  256
#define W_  256
#define INF_ 1e4f

typedef __attribute__((ext_vector_type(2))) float v2f;
typedef __attribute__((ext_vector_type(8))) float v8f;

// ---------------------------------------------------------------------------
// Kernel 0: init tmin slot (ignore_index handling: tmin = target.min())
// ---------------------------------------------------------------------------
__global__ void ea_init_kernel(int* __restrict__ tmin_slot) {
    if (blockIdx.x == 0 && threadIdx.x == 0) *tmin_slot = INT_MAX;
}

__global__ void ea_tmin_kernel(const int* __restrict__ target,
                               int* __restrict__ tmin_slot, int n) {
    int i = blockIdx.x * blockDim.x + threadIdx.x;
    int stride = gridDim.x * blockDim.x;
    int local = INT_MAX;
    for (; i < n; i += stride) local = min(local, target[i]);
    atomicMin(tmin_slot, local);   // integer min: order-independent, deterministic
}

// ---------------------------------------------------------------------------
// Kernel 1: vertical 1-D column distance (two passes), writes g^2 for all
// four masks into ws laid out [B][H][4][W] so a row tile is contiguous 4KB.
// One thread per (b, w) column; loads coalesced across w.
// Pass 2 recovers the mask from fwd!=0 (fwd>=1 iff mask true), so preds and
// target are read exactly once.
// ---------------------------------------------------------------------------
__global__ __launch_bounds__(256)
void ea_col_edt_kernel(const float* __restrict__ preds,
                       const int* __restrict__ target,
                       float* __restrict__ g2,
                       const int* __restrict__ tmin_slot) {
    int tid = blockIdx.x * blockDim.x + threadIdx.x;
    if (tid >= B_ * W_) return;
    int b = tid / W_;
    int w = tid - b * W_;
    int tmin = *tmin_slot;

    const float* p0base = preds + ((size_t)b * C_ + 0) * (size_t)(H_ * W_) + w;
    const float* p1base = preds + ((size_t)b * C_ + 1) * (size_t)(H_ * W_) + w;
    const int*   tbase  = target + (size_t)b * (size_t)(H_ * W_) + w;
    float*       gbase  = g2 + (size_t)b * (size_t)(H_ * 4 * W_) + w;

    // top-down scan: d = mask ? d+1 : 0, starting at INF
    float fpp = INF_, fpn = INF_, fgp = INF_, fgn = INF_;
    for (int r = 0; r < H_; ++r) {
        float p0 = p0base[(size_t)r * W_];
        float p1 = p1base[(size_t)r * W_];
        int   t  = tbase[(size_t)r * W_];
        if (t == 255) t = tmin;
        float gtf = (float)t;
        bool pcm = p1 > p0;        // argmax(C=2): class 1 wins only if strictly greater
        bool gtp = gtf > 0.0f;     // gt > 0
        bool gtn = gtf == 0.0f;    // gt == 0 (kept independent for faithfulness)
        fpp = pcm ? fpp + 1.0f : 0.0f;
        fpn = pcm ? 0.0f : fpn + 1.0f;
        fgp = gtp ? fgp + 1.0f : 0.0f;
        fgn = gtn ? fgn + 1.0f : 0.0f;
        float* row = gbase + (size_t)r * (4 * W_);
        row[0 * W_] = fpp;
        row[1 * W_] = fpn;
        row[2 * W_] = fgp;
        row[3 * W_] = fgn;
    }
    // bottom-up scan; g = min(fwd, bwd); store g^2 in place
    float bpp = INF_, bpn = INF_, bgp = INF_, bgn = INF_;
    for (int r = H_ - 1; r >= 0; --r) {
        float* row = gbase + (size_t)r * (4 * W_);
        float f0 = row[0 * W_], f1 = row[1 * W_], f2 = row[2 * W_], f3 = row[3 * W_];
        bpp = (f0 != 0.0f) ? bpp + 1.0f : 0.0f;   // mask recovered: fwd!=0 <=> mask
        bpn = (f1 != 0.0f) ? bpn + 1.0f : 0.0f;
        bgp = (f2 != 0.0f) ? bgp + 1.0f : 0.0f;
        bgn = (f3 != 0.0f) ? bgn + 1.0f : 0.0f;
        float g0 = fminf(f0, bpp), g1 = fminf(f1, bpn);
        float g2v = fminf(f2, bgp), g3 = fminf(f3, bgn);
        row[0 * W_] = g0 * g0;
        row[1 * W_] = g1 * g1;
        row[2 * W_] = g2v * g2v;
        row[3 * W_] = g3 * g3;
    }
}

// ---------------------------------------------------------------------------
// Kernel 2: horizontal min-plus pass + per-pixel loss + block partial sum.
// One workgroup (256 threads) per (b, i) row. The 4KB row tile (4 masks x
// 256 floats, contiguous in memory) is staged into LDS with the CDNA5 async
// DMA path: each thread issues one global_load_async_to_lds_b128, then
// s_wait_asynccnt 0 + barrier.
// ---------------------------------------------------------------------------
__global__ __launch_bounds__(256)
void ea_row_edt_loss_kernel(const float* __restrict__ preds,
                            const int* __restrict__ target,
                            const float* __restrict__ g2,
                            const int* __restrict__ tmin_slot,
                            float* __restrict__ partials) {
    __shared__ float s[4 * W_];     // 4KB row tile: [mask][k]
    __shared__ float red[256];

    int blk = blockIdx.x;           // blk = b*H + i
    int b = blk / H_;
    int i = blk - b * H_;
    int j = threadIdx.x;

    // Async DMA the row tile into LDS (16 bytes per lane covers 4KB).
    const float* row_base = g2 + (size_t)blk * (4 * W_);
    unsigned long long gaddr =
        (unsigned long long)(uintptr_t)row_base + (unsigned long long)j * 16ull;
    unsigned int laddr = (unsigned int)(uintptr_t)(&s[0]) + (unsigned int)j * 16u;
    asm volatile("global_load_async_to_lds_b128 %0, %1, off"
                 :: "v"(laddr), "v"(gaddr) : "memory");
    asm volatile("s_wait_asynccnt 0x0" ::: "memory");
    __syncthreads();

    // d2[j] = min_k (g2[k] + (j-k)^2) for each of the 4 masks.
    float m0 = 3.0e38f, m1 = 3.0e38f, m2 = 3.0e38f, m3 = 3.0e38f;
    #pragma unroll 8
    for (int k = 0; k < W_; ++k) {
        float diff = (float)(j - k);
        float d2 = diff * diff;
        m0 = fminf(m0, s[0 * W_ + k] + d2);
        m1 = fminf(m1, s[1 * W_ + k] + d2);
        m2 = fminf(m2, s[2 * W_ + k] + d2);
        m3 = fminf(m3, s[3 * W_ + k] + d2);
    }
    float pc_dist = sqrtf(m0) + sqrtf(m1);   // edt(pc>0) + edt(pc==0)
    float gt_dist = sqrtf(m2) + sqrtf(m3);   // edt(gt>0) + edt(gt==0)

    // per-pixel loss
    size_t pix = ((size_t)b * H_ + i) * W_ + j;
    float p0 = preds[((size_t)b * C_ + 0) * (size_t)(H_ * W_) + (size_t)i * W_ + j];
    float p1 = preds[((size_t)b * C_ + 1) * (size_t)(H_ * W_) + (size_t)i * W_ + j];
    int t = target[pix];
    int tmin = *tmin_slot;
    if (t == 255) t = tmin;
    float gtf = (float)t;
    float pc = (p1 > p0) ? 1.0f : 0.0f;
    float pred_error = fabsf(gtf - pc);
    float dist = sqrtf(pc_dist * pc_dist + gt_dist * gt_dist);
    float v = sqrtf(pred_error * dist + 1e-9f);

    // deterministic fixed-shape tree reduce
    red[j] = v;
    __syncthreads();
    for (int off = 128; off > 0; off >>= 1) {
        if (j < off) red[j] += red[j + off];
        __syncthreads();
    }
    if (j == 0) partials[blk] = red[0];
}

// ---------------------------------------------------------------------------
// Kernel 3: final reduction of 2048 partials via V_WMMA_F32_16X16X4_F32.
// A = all-ones => D[i,j] = sum_k B[k,j] + C[i,j]; the lane->(k,j) mapping is
// a bijection onto the 4x16 B matrix, and addition is commutative, so
// accumulating 64 floats per WMMA into C yields an exact f32 column-sum.
// All D rows are identical; total = sum of D[0, 0..15] (lanes 0..15, VGPR0).
// Single wave32 => EXEC all ones during the WMMAs.
// ---------------------------------------------------------------------------
__global__ __launch_bounds__(32)
void ea_final_reduce_kernel(const float* __restrict__ partials,
                            float* __restrict__ out) {
    int lane = threadIdx.x;
    v2f a; a.x = 1.0f; a.y = 1.0f;     // ones A-matrix fragment
    v8f c = {};
    #pragma unroll 4
    for (int it = 0; it < (B_ * H_) / 64; ++it) {   // 2048 / 64 = 32 chunks
        v2f bv;
        bv.x = partials[it * 64 + lane];
        bv.y = partials[it * 64 + 32 + lane];
        c = __builtin_amdgcn_wmma_f32_16x16x4_f32(
                /*neg_a=*/false, a, /*neg_b=*/false, bv,
                /*c_mod=*/(short)0, c, /*reuse_a=*/false, /*reuse_b=*/false);
    }
    __shared__ float tmp[16];
    if (lane < 16) tmp[lane] = c[0];   // row 0, cols 0..15
    __syncthreads();
    if (lane == 0) {
        float ssum = 0.0f;
        for (int q = 0; q < 16; ++q) ssum += tmp[q];
        out[0] = ssum / (float)(B_ * H_ * W_);
    }
}

// ---------------------------------------------------------------------------
// Launch
// ---------------------------------------------------------------------------
extern "C" void kernel_launch(void* const* d_in, const int* in_sizes, int n_in,
                              void* d_out, int out_size, void* d_ws, size_t ws_size,
                              hipStream_t stream) {
    (void)in_sizes; (void)n_in; (void)out_size; (void)ws_size;
    const float* preds  = (const float*)d_in[0];
    const int*   target = (const int*)d_in[1];
    float*       out    = (float*)d_out;

    // ws layout: g2[B][H][4][W] floats (8MB) | partials[B*H] | tmin (int)
    float* g2       = (float*)d_ws;
    float* partials = g2 + (size_t)B_ * H_ * 4 * W_;
    int*   tmin_slot = (int*)(partials + B_ * H_);

    ea_init_kernel<<<1, 1, 0, stream>>>(tmin_slot);
    ea_tmin_kernel<<<8, 256, 0, stream>>>(target, tmin_slot, B_ * H_ * W_);
    ea_col_edt_kernel<<<(B_ * W_ + 255) / 256, 256, 0, stream>>>(preds, target, g2, tmin_slot);
    ea_row_edt_loss_kernel<<<B_ * H_, 256, 0, stream>>>(preds, target, g2, tmin_slot, partials);
    ea_final_reduce_kernel<<<1, 32, 0, stream>>>(partials, out);
}